// BM3D_74354473828769
// MI455X (gfx1250) — compile-verified
//
#include <hip/hip_runtime.h>
#include <hip/hip_bf16.h>

// ---------------------------------------------------------------------------
// BM3D (sigma=0.8) on 256x256, fp32 end-to-end, using CDNA5 wave32 WMMA
// (v_wmma_f32_16x16x4_f32) for every matrix product:
//   - 2D DCT-16 of all 240x240 blocks
//   - K=32 / K=64 group-axis DCT forward+inverse (D32 / D64)
//   - per-patch 2D inverse DCT-16
// Aggregation uses global_atomic_add_f32. All dctA/dctB tensors (59MB each)
// are L2-resident on MI455X (192MB L2), so the candidate re-reads are cheap.
// ---------------------------------------------------------------------------

typedef float v2f __attribute__((ext_vector_type(2)));
typedef float v8f __attribute__((ext_vector_type(8)));

#define IMH 256
#define IMW 256
#define BS 16
#define NB 240              // IMH - BS
#define NCAND 576           // (WS-BS+1)^2 = 24*24
#define PI_F 3.14159265358979f

// workspace layout (float offsets)
#define OFF_D16   ((size_t)0)
#define OFF_D32   ((size_t)256)
#define OFF_D64   ((size_t)1280)
#define OFF_KAIS  ((size_t)5376)
#define OFF_ACC   ((size_t)5632)
#define OFF_WT    (OFF_ACC + 65536)
#define OFF_BASIC (OFF_WT + 65536)
#define OFF_DCTN  (OFF_BASIC + 65536)
#define OFF_DCTB  (OFF_DCTN + (size_t)NB * NB * 256)
// total: OFF_DCTB + 14,745,600 floats  ~= 118.8 MB  (assumes ws_size >= that)

__device__ __forceinline__ v8f wmma4(v2f a, v2f b, v8f c) {
    // D(16x16,f32) += A(16x4,f32) * B(4x16,f32)
    return __builtin_amdgcn_wmma_f32_16x16x4_f32(
        /*neg_a=*/false, a, /*neg_b=*/false, b,
        /*c_mod=*/(short)0, c, /*reuse_a=*/false, /*reuse_b=*/false);
}

// ---------------------------------------------------------------------------
// constants: DCT matrices D16/D32/D64 (row k, col m) and Kaiser(16,2) outer
// ---------------------------------------------------------------------------
__device__ __forceinline__ float dct_entry(int k, int m, int n) {
    float v = cosf(PI_F * (float)((2 * m + 1) * k) / (2.0f * (float)n)) *
              sqrtf(2.0f / (float)n);
    if (k == 0) v *= 0.70710678118654752f;
    return v;
}

__device__ __forceinline__ float bessel_i0f(float x) {
    float s = 1.f, t = 1.f, x2 = 0.25f * x * x;
    for (int i = 1; i <= 30; ++i) { t *= x2 / ((float)i * (float)i); s += t; }
    return s;
}

__global__ void init_consts_kernel(float* ws) {
    int t = blockIdx.x * blockDim.x + threadIdx.x;   // 4096 threads
    if (t < 256)  ws[OFF_D16 + t] = dct_entry(t >> 4, t & 15, 16);
    if (t < 1024) ws[OFF_D32 + t] = dct_entry(t >> 5, t & 31, 32);
    if (t < 4096) ws[OFF_D64 + t] = dct_entry(t >> 6, t & 63, 64);
    if (t < 256) {
        int i = t >> 4, j = t & 15;
        const float al = 7.5f, inv_i0b = 1.0f / bessel_i0f(2.0f);
        float xi = ((float)i - al) / al, xj = ((float)j - al) / al;
        float wi = bessel_i0f(2.0f * sqrtf(fmaxf(0.f, 1.f - xi * xi))) * inv_i0b;
        float wj = bessel_i0f(2.0f * sqrtf(fmaxf(0.f, 1.f - xj * xj))) * inv_i0b;
        ws[OFF_KAIS + t] = wi * wj;
    }
}

// ---------------------------------------------------------------------------
// 2D DCT-16 of every 16x16 block: out[by*240+bx][u*16+v] = (D16 P D16^T)[u][v]
// one wave per block, 8 waves / workgroup, all via WMMA
// ---------------------------------------------------------------------------
__global__ __launch_bounds__(256)
void dct_blocks_kernel(const float* img, const float* ws, float* out) {
    const float* D16m = ws + OFF_D16;
    __shared__ float wscr[8][256];
    const int wave = threadIdx.x >> 5, lane = threadIdx.x & 31;
    const int lm = lane & 15, hl = lane >> 4;
    const int blockId = blockIdx.x * 8 + wave;
    const int by = blockId / NB, bx = blockId % NB;

    // T = D16 @ P
    v8f t = {};
    for (int c = 0; c < 4; ++c) {
        int k0 = 4 * c + 2 * hl;
        v2f a, b;
        a.x = D16m[lm * 16 + k0];     a.y = D16m[lm * 16 + k0 + 1];
        b.x = img[(by + k0) * IMW + bx + lm];
        b.y = img[(by + k0 + 1) * IMW + bx + lm];
        t = wmma4(a, b, t);
    }
    for (int r = 0; r < 8; ++r) wscr[wave][(r + 8 * hl) * 16 + lm] = t[r];
    // R = T @ D16^T   (B[k][n] = D16[n][k], same element access as D16 A-load)
    v8f res = {};
    for (int c = 0; c < 4; ++c) {
        int k0 = 4 * c + 2 * hl;
        v2f a, b;
        a.x = wscr[wave][lm * 16 + k0]; a.y = wscr[wave][lm * 16 + k0 + 1];
        b.x = D16m[lm * 16 + k0];       b.y = D16m[lm * 16 + k0 + 1];
        res = wmma4(a, b, res);
    }
    float* o = out + (size_t)blockId * 256;
    for (int r = 0; r < 8; ++r) o[(r + 8 * hl) * 16 + lm] = res[r];
}

// ---------------------------------------------------------------------------
// helpers shared by step1/step2
// ---------------------------------------------------------------------------
__device__ __forceinline__ void grid_coords(int idx, int sp, int& r, int& t) {
    r = sp * idx; if (r > NB - 1) r = NB - 1;          // min(sp*i, 239)
    t = r - 11; if (t < 0) t = 0;                      // sh = int((16-39)/2) = -11
    if (t + 39 >= IMH) t = IMH - 1 - 39;               // 216
}

__device__ __forceinline__ void compute_dists(const float* dctbase, const float* ref,
                                              float* dist, int ti, int tj,
                                              int wave, int lane) {
    for (int cand = wave; cand < NCAND; cand += 8) {
        int oy = cand / 24, ox = cand - oy * 24;
        const float* c = dctbase + (size_t)((ti + oy) * NB + (tj + ox)) * 256;
        float s = 0.f;
        for (int e = lane; e < 256; e += 32) { float d = c[e] - ref[e]; s += d * d; }
        for (int off = 16; off > 0; off >>= 1) s += __shfl_xor(s, off, 32);
        if (lane == 0) dist[cand] = s;
    }
}

// iterative argmin (ascending distance, lowest index on ties == top_k order)
__device__ __forceinline__ void select_topk(float* dist, int* sel, int K,
                                            int wave, int lane) {
    if (wave == 0) {
        for (int k = 0; k < K; ++k) {
            float best = 3.0e38f; int bidx = 1 << 20;
            for (int c = lane; c < NCAND; c += 32) {
                float d = dist[c];
                if (d < best || (d == best && c < bidx)) { best = d; bidx = c; }
            }
            for (int off = 16; off > 0; off >>= 1) {
                float od = __shfl_xor(best, off, 32);
                int   oi = __shfl_xor(bidx, off, 32);
                if (od < best || (od == best && oi < bidx)) { best = od; bidx = oi; }
            }
            if (lane == 0) { sel[k] = bidx; dist[bidx] = 3.0e38f; }
        }
    }
}

// ---------------------------------------------------------------------------
// step1: one workgroup per (i,j) reference block; grid 82x82
// ---------------------------------------------------------------------------
__global__ __launch_bounds__(256)
void step1_kernel(float* ws) {
    const float* dctN = ws + OFF_DCTN;
    const float* D32m = ws + OFF_D32;
    const float* D16m = ws + OFF_D16;
    const float* KAIS = ws + OFF_KAIS;
    float* acc = ws + OFF_ACC;
    float* wt  = ws + OFF_WT;

    extern __shared__ float smem[];
    float* ref  = smem;                 // 256
    float* dist = smem + 256;           // 576
    float* wscr = smem + 832;           // 8*256
    float* grp  = smem + 2880;          // 32*256
    float* tbuf = smem + 11072;         // 32*256
    int*   sel  = (int*)(smem + 19264); // 32
    int*   ssy  = sel + 32;
    int*   ssx  = sel + 64;
    int*   sbase= sel + 96;
    int*   scnt = sel + 128;

    const int tid = threadIdx.x;
    const int wave = tid >> 5, lane = tid & 31, lm = lane & 15, hl = lane >> 4;
    int ri, ti, rj, tj;
    grid_coords(blockIdx.y, 3, ri, ti);
    grid_coords(blockIdx.x, 3, rj, tj);

    ref[tid] = dctN[(size_t)(ri * NB + rj) * 256 + tid];
    if (tid == 0) *scnt = 0;
    __syncthreads();

    compute_dists(dctN, ref, dist, ti, tj, wave, lane);
    __syncthreads();
    select_topk(dist, sel, 32, wave, lane);
    __syncthreads();

    if (tid < 32) {
        int c = sel[tid]; int oy = c / 24, ox = c - oy * 24;
        ssy[tid] = ti + oy; ssx[tid] = tj + ox;
        sbase[tid] = (ssy[tid] * NB + ssx[tid]) * 256;
    }
    __syncthreads();

    for (int x = tid; x < 32 * 256; x += 256)
        grp[x] = dctN[sbase[x >> 8] + (x & 255)];
    __syncthreads();

    // forward: t = D32 @ grp (32x256), hard-threshold, count nonzeros
    const float thre = 2.7f * 0.8f;
    int myc = 0;
    for (int q = 0; q < 4; ++q) {
        int tile = wave * 4 + q, kt = tile >> 4, et = tile & 15;
        v8f a8 = {};
        for (int c = 0; c < 8; ++c) {
            int k0 = 4 * c + 2 * hl;
            v2f a, b;
            a.x = D32m[(16 * kt + lm) * 32 + k0]; a.y = D32m[(16 * kt + lm) * 32 + k0 + 1];
            b.x = grp[k0 * 256 + 16 * et + lm];   b.y = grp[(k0 + 1) * 256 + 16 * et + lm];
            a8 = wmma4(a, b, a8);
        }
        for (int r = 0; r < 8; ++r) {
            float v = a8[r];
            v = (fabsf(v) < thre) ? 0.f : v;
            myc += (v != 0.f) ? 1 : 0;
            tbuf[(16 * kt + r + 8 * hl) * 256 + 16 * et + lm] = v;
        }
    }
    atomicAdd(scnt, myc);
    __syncthreads();
    float cntf = (float)(*scnt);
    float bw = (cntf < 1.f) ? 1.f : 1.f / (0.64f * cntf);

    // inverse: grp = D32^T @ tbuf
    for (int q = 0; q < 4; ++q) {
        int tile = wave * 4 + q, mt = tile >> 4, et = tile & 15;
        v8f a8 = {};
        for (int c = 0; c < 8; ++c) {
            int k0 = 4 * c + 2 * hl;
            v2f a, b;
            a.x = D32m[k0 * 32 + 16 * mt + lm];   a.y = D32m[(k0 + 1) * 32 + 16 * mt + lm];
            b.x = tbuf[k0 * 256 + 16 * et + lm];  b.y = tbuf[(k0 + 1) * 256 + 16 * et + lm];
            a8 = wmma4(a, b, a8);
        }
        for (int r = 0; r < 8; ++r)
            grp[(16 * mt + r + 8 * hl) * 256 + 16 * et + lm] = a8[r];
    }
    __syncthreads();

    // per-patch 2D inverse DCT + weighted scatter
    for (int q = 0; q < 4; ++q) {
        int m = wave * 4 + q;
        const float* blk = grp + m * 256;
        float* wsc = wscr + wave * 256;
        v8f t8 = {};
        for (int c = 0; c < 4; ++c) {
            int k0 = 4 * c + 2 * hl;
            v2f a, b;
            a.x = D16m[k0 * 16 + lm]; a.y = D16m[(k0 + 1) * 16 + lm];
            b.x = blk[k0 * 16 + lm];  b.y = blk[(k0 + 1) * 16 + lm];
            t8 = wmma4(a, b, t8);                 // T = D16^T @ blk
        }
        for (int r = 0; r < 8; ++r) wsc[(r + 8 * hl) * 16 + lm] = t8[r];
        v8f r8 = {};
        for (int c = 0; c < 4; ++c) {
            int k0 = 4 * c + 2 * hl;
            v2f a, b;
            a.x = wsc[lm * 16 + k0];  a.y = wsc[lm * 16 + k0 + 1];
            b.x = D16m[k0 * 16 + lm]; b.y = D16m[(k0 + 1) * 16 + lm];
            r8 = wmma4(a, b, r8);                 // sp = T @ D16
        }
        int py = ssy[m], px = ssx[m];
        for (int r = 0; r < 8; ++r) {
            int row = r + 8 * hl, col = lm;
            float wp = bw * KAIS[row * 16 + col];
            int g = (py + row) * IMW + px + col;
            atomicAdd(acc + g, wp * r8[r]);
            atomicAdd(wt + g, wp);
        }
    }
}

// ---------------------------------------------------------------------------
// step2: Wiener pass; one workgroup per (i,j); grid 62x62; K2=64
// ---------------------------------------------------------------------------
__global__ __launch_bounds__(256)
void step2_kernel(float* ws) {
    const float* dctB = ws + OFF_DCTB;
    const float* dctN = ws + OFF_DCTN;
    const float* D64m = ws + OFF_D64;
    const float* D16m = ws + OFF_D16;
    const float* KAIS = ws + OFF_KAIS;
    float* acc = ws + OFF_ACC;
    float* wt  = ws + OFF_WT;

    extern __shared__ float smem[];
    float* ref  = smem;                 // 256
    float* dist = smem + 256;           // 576
    float* wscr = smem + 832;           // 8*256
    float* grp  = smem + 2880;          // 64*256
    float* tbuf = smem + 19264;         // 64*256
    int*   sel  = (int*)(smem + 35648); // 64
    int*   ssy  = sel + 64;
    int*   ssx  = sel + 128;
    int*   sbase= sel + 192;
    float* swsum= (float*)(sel + 256);

    const int tid = threadIdx.x;
    const int wave = tid >> 5, lane = tid & 31, lm = lane & 15, hl = lane >> 4;
    int ri, ti, rj, tj;
    grid_coords(blockIdx.y, 4, ri, ti);
    grid_coords(blockIdx.x, 4, rj, tj);

    ref[tid] = dctB[(size_t)(ri * NB + rj) * 256 + tid];
    if (tid == 0) *swsum = 0.f;
    __syncthreads();

    compute_dists(dctB, ref, dist, ti, tj, wave, lane);
    __syncthreads();
    select_topk(dist, sel, 64, wave, lane);
    __syncthreads();

    if (tid < 64) {
        int c = sel[tid]; int oy = c / 24, ox = c - oy * 24;
        ssy[tid] = ti + oy; ssx[tid] = tj + ox;
        sbase[tid] = (ssy[tid] * NB + ssx[tid]) * 256;
    }
    __syncthreads();

    for (int x = tid; x < 64 * 256; x += 256)
        grp[x] = dctB[sbase[x >> 8] + (x & 255)];
    __syncthreads();

    // tB = D64 @ grpB (LDS), tN = D64 @ grpN (direct from global dctN, L2-hot)
    const float s2 = 0.64f, coef = 1.0f / 64.0f;
    float mysum = 0.f;
    for (int q = 0; q < 8; ++q) {
        int tile = wave * 8 + q, kt = tile >> 4, et = tile & 15;
        v8f tb = {}, tn = {};
        for (int c = 0; c < 16; ++c) {
            int k0 = 4 * c + 2 * hl;
            v2f a, bb, bn;
            a.x = D64m[(16 * kt + lm) * 64 + k0]; a.y = D64m[(16 * kt + lm) * 64 + k0 + 1];
            bb.x = grp[k0 * 256 + 16 * et + lm];  bb.y = grp[(k0 + 1) * 256 + 16 * et + lm];
            tb = wmma4(a, bb, tb);
            bn.x = dctN[sbase[k0] + 16 * et + lm];
            bn.y = dctN[sbase[k0 + 1] + 16 * et + lm];
            tn = wmma4(a, bn, tn);
        }
        for (int r = 0; r < 8; ++r) {
            float vb = tb[r];
            float val = vb * vb * coef;
            val = val / (val + s2);
            mysum += val;
            tbuf[(16 * kt + r + 8 * hl) * 256 + 16 * et + lm] = tn[r] * val;
        }
    }
    atomicAdd(swsum, mysum);
    __syncthreads();
    float wsum = *swsum;
    float ww = (wsum > 0.f) ? 1.f / (s2 * wsum) : 1.f;

    // inverse: grp = D64^T @ tbuf
    for (int q = 0; q < 8; ++q) {
        int tile = wave * 8 + q, mt = tile >> 4, et = tile & 15;
        v8f a8 = {};
        for (int c = 0; c < 16; ++c) {
            int k0 = 4 * c + 2 * hl;
            v2f a, b;
            a.x = D64m[k0 * 64 + 16 * mt + lm];  a.y = D64m[(k0 + 1) * 64 + 16 * mt + lm];
            b.x = tbuf[k0 * 256 + 16 * et + lm]; b.y = tbuf[(k0 + 1) * 256 + 16 * et + lm];
            a8 = wmma4(a, b, a8);
        }
        for (int r = 0; r < 8; ++r)
            grp[(16 * mt + r + 8 * hl) * 256 + 16 * et + lm] = a8[r];
    }
    __syncthreads();

    // per-patch 2D inverse DCT + weighted scatter
    for (int q = 0; q < 8; ++q) {
        int m = wave * 8 + q;
        const float* blk = grp + m * 256;
        float* wsc = wscr + wave * 256;
        v8f t8 = {};
        for (int c = 0; c < 4; ++c) {
            int k0 = 4 * c + 2 * hl;
            v2f a, b;
            a.x = D16m[k0 * 16 + lm]; a.y = D16m[(k0 + 1) * 16 + lm];
            b.x = blk[k0 * 16 + lm];  b.y = blk[(k0 + 1) * 16 + lm];
            t8 = wmma4(a, b, t8);
        }
        for (int r = 0; r < 8; ++r) wsc[(r + 8 * hl) * 16 + lm] = t8[r];
        v8f r8 = {};
        for (int c = 0; c < 4; ++c) {
            int k0 = 4 * c + 2 * hl;
            v2f a, b;
            a.x = wsc[lm * 16 + k0];  a.y = wsc[lm * 16 + k0 + 1];
            b.x = D16m[k0 * 16 + lm]; b.y = D16m[(k0 + 1) * 16 + lm];
            r8 = wmma4(a, b, r8);
        }
        int py = ssy[m], px = ssx[m];
        for (int r = 0; r < 8; ++r) {
            int row = r + 8 * hl, col = lm;
            float wp = ww * KAIS[row * 16 + col];
            int g = (py + row) * IMW + px + col;
            atomicAdd(acc + g, wp * r8[r]);
            atomicAdd(wt + g, wp);
        }
    }
}

// ---------------------------------------------------------------------------
__global__ void normalize_kernel(const float* acc, const float* wt, float* out) {
    int t = blockIdx.x * 256 + threadIdx.x;
    float w = wt[t];
    out[t] = acc[t] / ((w == 0.f) ? 1.f : w);
}

// ---------------------------------------------------------------------------
extern "C" void kernel_launch(void* const* d_in, const int* in_sizes, int n_in,
                              void* d_out, int out_size, void* d_ws, size_t ws_size,
                              hipStream_t stream) {
    (void)in_sizes; (void)n_in; (void)out_size; (void)ws_size;
    const float* x = (const float*)d_in[0];     // (1,1,256,256) f32
    float* out = (float*)d_out;                 // 65536 f32
    float* ws = (float*)d_ws;                   // needs ~119 MB

    const size_t sh1 = 19400 * sizeof(float);   // step1 dynamic LDS (~78 KB)
    const size_t sh2 = 35912 * sizeof(float);   // step2 dynamic LDS (~141 KB)

    init_consts_kernel<<<16, 256, 0, stream>>>(ws);
    hipMemsetAsync(ws + OFF_ACC, 0, 2 * 65536 * sizeof(float), stream);

    // ---- step 1 (hard threshold) ----
    dct_blocks_kernel<<<(NB * NB) / 8, 256, 0, stream>>>(x, ws, ws + OFF_DCTN);
    dim3 g1(82, 82);
    step1_kernel<<<g1, 256, sh1, stream>>>(ws);
    normalize_kernel<<<256, 256, 0, stream>>>(ws + OFF_ACC, ws + OFF_WT, ws + OFF_BASIC);

    // ---- step 2 (Wiener) ----
    dct_blocks_kernel<<<(NB * NB) / 8, 256, 0, stream>>>(ws + OFF_BASIC, ws, ws + OFF_DCTB);
    hipMemsetAsync(ws + OFF_ACC, 0, 2 * 65536 * sizeof(float), stream);
    dim3 g2(62, 62);
    step2_kernel<<<g2, 256, sh2, stream>>>(ws);
    normalize_kernel<<<256, 256, 0, stream>>>(ws + OFF_ACC, ws + OFF_WT, out);
}